// iSogCLR_New_v2_Loss_9972914061430
// MI455X (gfx1250) — compile-verified
//
#include <hip/hip_runtime.h>
#include <stdint.h>

typedef __attribute__((ext_vector_type(2))) float v2f;
typedef __attribute__((ext_vector_type(8))) float v8f;

#define GAMMA_C 0.8f
#define TAU_MIN_C 0.005f
#define TAU_MAX_C 1.0f
#define RHO_C 6.0f
#define ETA_C 0.01f
#define BETA_U_C 0.9f
#define EPS_C 1e-8f

#define KC 32    // K panel in LDS
#define MT 64    // block tile M (4 wave rows x 16)
#define NTB 128  // block tile N (2 wave cols x 4 accumulators x 16)

// ---------------------------------------------------------------- reductions
__device__ __forceinline__ float blk_reduce_sum(float v, float* red, int tid) {
  red[tid] = v; __syncthreads();
  for (int s = 128; s > 0; s >>= 1) {
    if (tid < s) red[tid] += red[tid + s];
    __syncthreads();
  }
  float r = red[0]; __syncthreads();
  return r;
}

__device__ __forceinline__ float blk_reduce_max(float v, float* red, int tid) {
  red[tid] = v; __syncthreads();
  for (int s = 128; s > 0; s >>= 1) {
    if (tid < s) red[tid] = fmaxf(red[tid], red[tid + s]);
    __syncthreads();
  }
  float r = red[0]; __syncthreads();
  return r;
}

// ---------------------------------------------------------------- bulk copy
// dst may be misaligned by <16B (out+7); align head, b128 body, scalar tail.
__global__ __launch_bounds__(256) void copy_f32(const float* __restrict__ src,
                                                float* __restrict__ dst,
                                                long long n) {
  long long i = (long long)blockIdx.x * blockDim.x + threadIdx.x;
  long long stride = (long long)gridDim.x * blockDim.x;
  long long head = (long long)(((16u - ((uint32_t)(uintptr_t)dst & 15u)) & 15u) >> 2);
  if (head > n) head = n;
  for (long long k = i; k < head; k += stride) dst[k] = src[k];
  const float* s2 = src + head;
  float* d2 = dst + head;
  long long nv = (n - head) >> 2;
  for (long long k = i; k < nv; k += stride) {
    const float* p = s2 + 4 * k;
    float4 v = make_float4(p[0], p[1], p[2], p[3]);
    ((float4*)d2)[k] = v;  // d2 is 16B aligned
  }
  for (long long k = head + nv * 4 + i; k < n; k += stride) dst[k] = src[k];
}

// ---------------------------------------------------------------- WMMA GEMM
// sim[M,N] = img[M,K] * txt[N,K]^T  (fp32, V_WMMA_F32_16X16X4_F32)
// 8 waves: 4 wave-rows x 2 wave-cols; each wave owns a 16x64 strip
// (4 independent accumulators -> 4 pipelined WMMA dependence chains).
__global__ __launch_bounds__(256) void sim_gemm_wmma(const float* __restrict__ img,
                                                     const float* __restrict__ txt,
                                                     float* __restrict__ sim,
                                                     int Bsz, int Dd) {
  __shared__ __align__(16) float As[MT][KC];    // 8 KB
  __shared__ __align__(16) float Bs[NTB][KC];   // 16 KB
  const int tid  = threadIdx.x;
  const int lane = tid & 31;
  const int wid  = tid >> 5;
  const int m_blk = blockIdx.y * MT;
  const int n_blk = blockIdx.x * NTB;
  const int m0 = (wid >> 1) * 16;   // wave row tile within block
  const int n0 = (wid & 1) * 64;    // wave col strip within block
  // fragment lane decomposition (ISA 16x4 f32 A layout; B mirrored)
  const int fm = lane & 15;           // M (for A) / N (for B)
  const int fk = (lane >> 4) << 1;    // K base: lanes 16-31 hold K+2,K+3
  // staging decomposition: 8 float4 slots per 32-float row
  const int st_r  = tid >> 3;         // row 0..31
  const int st_c4 = (tid & 7) << 2;   // float4 col offset

  v8f acc[4] = {v8f{}, v8f{}, v8f{}, v8f{}};

  for (int k0 = 0; k0 < Dd; k0 += KC) {
    // stage A: 64x32 floats -> exactly 2 float4 per thread (branch-free)
#pragma unroll
    for (int u = 0; u < 2; ++u) {
      int r = st_r + 32 * u;
      *(float4*)&As[r][st_c4] =
          *(const float4*)(img + (size_t)(m_blk + r) * Dd + k0 + st_c4);
    }
    // stage B: 128x32 floats -> exactly 4 float4 per thread
#pragma unroll
    for (int u = 0; u < 4; ++u) {
      int r = st_r + 32 * u;
      *(float4*)&Bs[r][st_c4] =
          *(const float4*)(txt + (size_t)(n_blk + r) * Dd + k0 + st_c4);
    }
    if (k0 + KC < Dd) {  // uniform guard: prefetch next K panel
#pragma unroll
      for (int u = 0; u < 2; ++u)
        __builtin_prefetch(img + (size_t)(m_blk + st_r + 32 * u) * Dd + k0 + KC + st_c4, 0, 1);
#pragma unroll
      for (int u = 0; u < 4; ++u)
        __builtin_prefetch(txt + (size_t)(n_blk + st_r + 32 * u) * Dd + k0 + KC + st_c4, 0, 1);
    }
    __syncthreads();
#pragma unroll
    for (int kk = 0; kk < KC; kk += 4) {
      v2f a;
      a.x = As[m0 + fm][kk + fk];
      a.y = As[m0 + fm][kk + fk + 1];
#pragma unroll
      for (int t = 0; t < 4; ++t) {
        v2f b;
        b.x = Bs[n0 + 16 * t + fm][kk + fk];
        b.y = Bs[n0 + 16 * t + fm][kk + fk + 1];
        acc[t] = __builtin_amdgcn_wmma_f32_16x16x4_f32(false, a, false, b,
                                                       (short)0, acc[t],
                                                       false, false);
      }
    }
    __syncthreads();
  }
  // C/D layout: VGPR r -> M = r + 8*(lane>>4), N = lane&15
  const int cn = lane & 15;
  const int cm = (lane >> 4) * 8;
#pragma unroll
  for (int t = 0; t < 4; ++t) {
#pragma unroll
    for (int r = 0; r < 8; ++r) {
      sim[(size_t)(m_blk + m0 + cm + r) * Bsz + (n_blk + n0 + 16 * t + cn)] =
          acc[t][r];
    }
  }
}

// ---------------------------------------------------------------- per-row/col stats
// rows (image): fixedStride=B, runStride=1 ; cols (text): fixedStride=1, runStride=B
__global__ __launch_bounds__(256) void side_kernel(
    const float* __restrict__ sim, const long long* __restrict__ ids,
    const float* __restrict__ s_in, const float* __restrict__ tau_in,
    const float* __restrict__ u_in, const float* __restrict__ b_in,
    float* __restrict__ s_out, float* __restrict__ tau_out,
    float* __restrict__ u_out, float* __restrict__ b_out,
    float* __restrict__ ws_loss, float* __restrict__ ws_grad,
    float* __restrict__ ws_taub, float* __restrict__ ws_oldb,
    const int* __restrict__ epoch_p,
    int Bsz, size_t fixedStride, size_t runStride) {
  __shared__ float red[256];
  const int tid = threadIdx.x;
  const int i = blockIdx.x;
  const size_t base = (size_t)i * fixedStride;
  const long long id = ids[i];
  const float tau = tau_in[id];
  const float inv_tau = 1.0f / tau;

  // pass 1: max over full row/col of sim/tau (diag included, matching reference)
  float lmax = -3.4e38f;
  for (int j = tid; j < Bsz; j += 256)
    lmax = fmaxf(lmax, sim[base + (size_t)j * runStride] * inv_tau);
  const float rowmax = blk_reduce_max(lmax, red, tid);
  const float b_old = b_in[id];
  const float b_new = fmaxf(rowmax, b_old);

  // pass 2: masked exp sums
  float se = 0.0f, ss = 0.0f, st = 0.0f;
  for (int j = tid; j < Bsz; j += 256) {
    float sv = sim[base + (size_t)j * runStride];
    float t = sv * inv_tau;
    float e = (j == i) ? 0.0f : __expf(t - b_new);
    se += e; ss += e * sv; st += e * t;
  }
  se = blk_reduce_sum(se, red, tid);
  ss = blk_reduce_sum(ss, red, tid);
  st = blk_reduce_sum(st, red, tid);

  if (tid == 0) {
    const float invBm1 = 1.0f / (float)(Bsz - 1);
    const float g = se * invBm1;
    const int ep = *epoch_p;
    const float s_new = (ep == 0)
        ? g
        : (1.0f - GAMMA_C) * s_in[id] * __expf(b_old - b_new) + GAMMA_C * g;
    const float inv_s = 1.0f / (s_new + EPS_C);
    const float loss = ss * inv_s * invBm1;
    const float grad = __logf(s_new) + b_new + RHO_C - st * inv_s * invBm1;
    const float ub = (1.0f - BETA_U_C) * u_in[id] + BETA_U_C * grad;
    const float tb = fminf(fmaxf(tau - ETA_C * ub, TAU_MIN_C), TAU_MAX_C);
    // scatter into global state (overwrites bulk copy)
    s_out[id] = s_new; b_out[id] = b_new; u_out[id] = ub; tau_out[id] = tb;
    // per-row partials for finalize
    ws_loss[i] = loss; ws_grad[i] = grad; ws_taub[i] = tb; ws_oldb[i] = b_old;
  }
}

// ---------------------------------------------------------------- finalize
__global__ __launch_bounds__(256) void finalize_kernel(
    const float* __restrict__ sim,
    const float* __restrict__ il_i, const float* __restrict__ gr_i,
    const float* __restrict__ tb_i, const float* __restrict__ ob_i,
    const float* __restrict__ il_t, const float* __restrict__ gr_t,
    const float* __restrict__ tb_t, const float* __restrict__ ob_t,
    const float* __restrict__ b_pos_p, const float* __restrict__ lamda_p,
    const float* __restrict__ r_p, const int* __restrict__ epoch_p,
    float* __restrict__ out, long long Nsz, int Bsz) {
  __shared__ float red[256];
  const int tid = threadIdx.x;
  const float lamda = *lamda_p;

  float sil = 0, stl = 0, sgi = 0, sgt = 0, sti = 0, stt = 0, sbi = 0, sbt = 0;
  float pmax = -3.4e38f;
  for (int i = tid; i < Bsz; i += 256) {
    sil += il_i[i]; stl += il_t[i];
    sgi += gr_i[i]; sgt += gr_t[i];
    sti += tb_i[i]; stt += tb_t[i];
    sbi += ob_i[i]; sbt += ob_t[i];
    float diag = sim[(size_t)i * Bsz + i];
    pmax = fmaxf(pmax, -2.0f * diag / lamda);
  }
  sil = blk_reduce_sum(sil, red, tid);
  stl = blk_reduce_sum(stl, red, tid);
  sgi = blk_reduce_sum(sgi, red, tid);
  sgt = blk_reduce_sum(sgt, red, tid);
  sti = blk_reduce_sum(sti, red, tid);
  stt = blk_reduce_sum(stt, red, tid);
  sbi = blk_reduce_sum(sbi, red, tid);
  sbt = blk_reduce_sum(sbt, red, tid);
  pmax = blk_reduce_max(pmax, red, tid);

  const float b_pos_old = *b_pos_p;
  const float b_pos_new = fmaxf(b_pos_old, pmax);

  float se = 0, sed = 0;
  for (int i = tid; i < Bsz; i += 256) {
    float diag = sim[(size_t)i * Bsz + i];
    float e = __expf(-2.0f * diag / lamda - b_pos_new);
    se += e; sed += e * diag;
  }
  se = blk_reduce_sum(se, red, tid);
  sed = blk_reduce_sum(sed, red, tid);

  if (tid == 0) {
    const float invB = 1.0f / (float)Bsz;
    const float mean_exp = se * invB;
    const int ep = *epoch_p;
    const float r_old = *r_p;
    const float r_new = (ep == 0)
        ? mean_exp
        : (1.0f - GAMMA_C) * r_old * __expf(b_pos_old - b_pos_new) + GAMMA_C * mean_exp;
    const float pos_mean = -2.0f * sed / r_new * invB;  // mean of -2*diag_w*diag
    out[0] = sil * invB + stl * invB + pos_mean;        // total_loss
    out[1] = sti * invB;                                // tau_I_b.mean
    out[2] = stt * invB;                                // tau_T_b.mean
    out[3] = sgi * invB;                                // grad_tau_img.mean
    out[4] = sgt * invB;                                // grad_tau_txt.mean
    out[5] = sbi * invB;                                // old_b_I.mean
    out[6] = sbt * invB;                                // old_b_T.mean
    out[7 + 8 * Nsz]     = b_pos_new;
    out[7 + 8 * Nsz + 1] = r_new;
  }
}

// ---------------------------------------------------------------- launch
extern "C" void kernel_launch(void* const* d_in, const int* in_sizes, int n_in,
                              void* d_out, int out_size, void* d_ws, size_t ws_size,
                              hipStream_t stream) {
  const float* img   = (const float*)d_in[0];
  const float* txt   = (const float*)d_in[1];
  const float* s_I   = (const float*)d_in[2];
  const float* s_T   = (const float*)d_in[3];
  const float* tau_I = (const float*)d_in[4];
  const float* tau_T = (const float*)d_in[5];
  const float* u_I   = (const float*)d_in[6];
  const float* u_T   = (const float*)d_in[7];
  const float* b_I   = (const float*)d_in[8];
  const float* b_T   = (const float*)d_in[9];
  const float* b_pos = (const float*)d_in[10];
  const float* lamda = (const float*)d_in[11];
  const float* r     = (const float*)d_in[12];
  const long long* image_ids = (const long long*)d_in[13];  // jnp.int64
  const long long* text_ids  = (const long long*)d_in[14];
  const int* epoch = (const int*)d_in[15];

  const long long Nsz = in_sizes[2];
  const int Bsz = in_sizes[13];
  const int Dd  = in_sizes[0] / Bsz;

  float* out = (float*)d_out;
  float* sim = (float*)d_ws;                       // B*B floats
  float* wsr = sim + (size_t)Bsz * Bsz;            // 8 * B per-row partials
  float* il_i = wsr + 0 * Bsz; float* gr_i = wsr + 1 * Bsz;
  float* tb_i = wsr + 2 * Bsz; float* ob_i = wsr + 3 * Bsz;
  float* il_t = wsr + 4 * Bsz; float* gr_t = wsr + 5 * Bsz;
  float* tb_t = wsr + 6 * Bsz; float* ob_t = wsr + 7 * Bsz;

  float* s_I_out   = out + 7;
  float* s_T_out   = s_I_out + Nsz;
  float* tau_I_out = s_T_out + Nsz;
  float* tau_T_out = tau_I_out + Nsz;
  float* u_I_out   = tau_T_out + Nsz;
  float* u_T_out   = u_I_out + Nsz;
  float* b_I_out   = u_T_out + Nsz;
  float* b_T_out   = b_I_out + Nsz;

  // 1) bulk state copies (HBM-bound part: ~186 MB total)
  const int cpB = 2048;
  copy_f32<<<cpB, 256, 0, stream>>>(s_I,   s_I_out,   Nsz);
  copy_f32<<<cpB, 256, 0, stream>>>(s_T,   s_T_out,   Nsz);
  copy_f32<<<cpB, 256, 0, stream>>>(tau_I, tau_I_out, Nsz);
  copy_f32<<<cpB, 256, 0, stream>>>(tau_T, tau_T_out, Nsz);
  copy_f32<<<cpB, 256, 0, stream>>>(u_I,   u_I_out,   Nsz);
  copy_f32<<<cpB, 256, 0, stream>>>(u_T,   u_T_out,   Nsz);
  copy_f32<<<cpB, 256, 0, stream>>>(b_I,   b_I_out,   Nsz);
  copy_f32<<<cpB, 256, 0, stream>>>(b_T,   b_T_out,   Nsz);

  // 2) sim = img @ txt^T via fp32 WMMA (64x128 block tile, 16x64 per wave)
  dim3 gg(Bsz / NTB, Bsz / MT);
  sim_gemm_wmma<<<gg, 256, 0, stream>>>(img, txt, sim, Bsz, Dd);

  // 3) image rows
  side_kernel<<<Bsz, 256, 0, stream>>>(sim, image_ids, s_I, tau_I, u_I, b_I,
      s_I_out, tau_I_out, u_I_out, b_I_out, il_i, gr_i, tb_i, ob_i,
      epoch, Bsz, (size_t)Bsz, (size_t)1);
  // 4) text cols
  side_kernel<<<Bsz, 256, 0, stream>>>(sim, text_ids, s_T, tau_T, u_T, b_T,
      s_T_out, tau_T_out, u_T_out, b_T_out, il_t, gr_t, tb_t, ob_t,
      epoch, Bsz, (size_t)1, (size_t)Bsz);

  // 5) scalars
  finalize_kernel<<<1, 256, 0, stream>>>(sim, il_i, gr_i, tb_i, ob_i,
      il_t, gr_t, tb_t, ob_t, b_pos, lamda, r, epoch, out, Nsz, Bsz);
}